// SegDET_41781441855750
// MI455X (gfx1250) — compile-verified
//
#include <hip/hip_runtime.h>
#include <cstdint>
#include <cstddef>

// ---------------------------------------------------------------------------
// Problem constants (match reference)
// ---------------------------------------------------------------------------
#define B_    8
#define Q_    900
#define NCLS  11
#define C_    256
#define NA    100     // NUM_ANCHOR
#define H_    200
#define W_    200
#define HW_   (H_ * W_)

typedef __attribute__((ext_vector_type(2))) float v2f;
typedef __attribute__((ext_vector_type(8))) float v8f;

// ---------------------------------------------------------------------------
// Async DMA staging of the 100 box-bound int4s into LDS (CDNA5 ASYNCcnt path).
// Threads 0..NA-1 each move 16 B: global_load_async_to_lds_b128 writes LDS
// directly (no VGPR round trip). the LDS address comes from the VDST VGPR; low 32 bits of a
// generic __shared__ pointer are the wave-relative LDS byte offset (ISA 10.2).
// ---------------------------------------------------------------------------
__device__ __forceinline__ void stage_bounds_async(int4* sb_lds,
                                                   const int4* __restrict__ gsrc) {
    if (threadIdx.x < NA) {
        unsigned lds_addr = (unsigned)(uintptr_t)sb_lds + threadIdx.x * 16u;
        unsigned goff     = threadIdx.x * 16u;
        asm volatile("global_load_async_to_lds_b128 %0, %1, %2"
                     :
                     : "v"(lds_addr), "v"(goff), "s"(gsrc)
                     : "memory");
    }
    asm volatile("s_wait_asynccnt 0" ::: "memory");
    __syncthreads();
}

// ---------------------------------------------------------------------------
// Kernel 1: scores[b,q] = max over non-background classes of softmax(logits)
// ---------------------------------------------------------------------------
__global__ void seg_scores_kernel(const float* __restrict__ logits,
                                  float* __restrict__ scores) {
    int t = blockIdx.x * blockDim.x + threadIdx.x;
    if (t >= B_ * Q_) return;
    const float* l = logits + (size_t)t * NCLS;
    float m = l[0];
#pragma unroll
    for (int i = 1; i < NCLS; ++i) m = fmaxf(m, l[i]);
    float denom = 0.f;
#pragma unroll
    for (int i = 0; i < NCLS; ++i) denom += __expf(l[i] - m);
    float m10 = l[0];
#pragma unroll
    for (int i = 1; i < NCLS - 1; ++i) m10 = fmaxf(m10, l[i]);  // exclude background
    scores[t] = __expf(m10 - m) / denom;
}

// ---------------------------------------------------------------------------
// Kernel 2: stable top-k (k=100) per batch via rank counting.
// rank(q) = #{j : s[j] > s[q]  or  (s[j]==s[q] and j<q)}  -> matches lax.top_k
// (ranks are all distinct, so exactly NA slots are written every call)
// ---------------------------------------------------------------------------
__global__ void seg_topk_kernel(const float* __restrict__ scores,
                                int* __restrict__ idx_sel) {
    __shared__ float s[Q_];
    const int b = blockIdx.x;
    const float* sc = scores + (size_t)b * Q_;
    for (int q = threadIdx.x; q < Q_; q += blockDim.x) s[q] = sc[q];
    __syncthreads();
    for (int q = threadIdx.x; q < Q_; q += blockDim.x) {
        const float v = s[q];
        int rank = 0;
        for (int j = 0; j < Q_; ++j) {
            const float u = s[j];
            rank += (u > v) || (u == v && j < q);
        }
        if (rank < NA) idx_sel[b * NA + rank] = q;
    }
}

// ---------------------------------------------------------------------------
// Kernel 3: decode selected boxes -> clipped integer BEV pixel bounds (int4).
// xyxy1 = (cx-w/2, cy-h/2, cx+w/2, cy+h/2, 1); coords = view[b] @ xyxy1
// ---------------------------------------------------------------------------
__global__ void seg_decode_kernel(const float* __restrict__ boxes,
                                  const float* __restrict__ view,
                                  const int* __restrict__ idx_sel,
                                  int4* __restrict__ bounds) {
    int t = blockIdx.x * blockDim.x + threadIdx.x;
    if (t >= B_ * NA) return;
    const int b = t / NA;
    const int q = idx_sel[t];
    const float* bx = boxes + ((size_t)b * Q_ + q) * 4;
    const float cx = bx[0], cy = bx[1];
    const float w = __expf(bx[2]), h = __expf(bx[3]);
    const float xy[5] = {cx - 0.5f * w, cy - 0.5f * h,
                         cx + 0.5f * w, cy + 0.5f * h, 1.0f};
    const float* V = view + (size_t)b * 25;
    int r[4];
#pragma unroll
    for (int i = 0; i < 4; ++i) {
        float acc = 0.f;
#pragma unroll
        for (int j = 0; j < 5; ++j) acc += V[i * 5 + j] * xy[j];
        int c = (int)acc;                       // truncation, like astype(int32)
        const int hi = (i & 1) ? H_ : W_;       // x clipped to [0,W], y to [0,H]
        r[i] = min(max(c, 0), hi);
    }
    bounds[t] = make_int4(r[0], r[1], r[2], r[3]);   // (x0, y0, x1, y1)
}

// ---------------------------------------------------------------------------
// Kernel 4: gather feats of selected queries -> fg[b*NA+n][c]  (coalesced)
// ---------------------------------------------------------------------------
__global__ void seg_gather_kernel(const float* __restrict__ feats,
                                  const int* __restrict__ idx_sel,
                                  float* __restrict__ fg) {
    const int bn = blockIdx.x;            // 0 .. B*NA-1
    const int b  = bn / NA;
    const int q  = idx_sel[bn];
    fg[(size_t)bn * C_ + threadIdx.x] =
        feats[((size_t)b * Q_ + q) * C_ + threadIdx.x];
}

// ---------------------------------------------------------------------------
// Kernel 5 (main): bev[b, p, c] = sum_n region(b,n,p) * fg(b,n,c)
// Per batch: GEMM  [HW x NA] (0/1, generated in-registers) x [NA x C].
// One wave = 16(M pixels) x 32(N channels) tile: the A operand (region
// indicators) is built once per K4-step and reused across 2 WMMAs.
// K = NA = 100 -> 25 x V_WMMA_F32_16X16X4_F32 steps per accumulator.
//
// A layout (ISA 32-bit A 16x4): lane<16 -> M=lane, K={k,k+1};
//                               lane>=16 -> M=lane-16, K={k+2,k+3}.
// B layout mirrored: lane<16 -> N=lane, rows K={k,k+1}; lane>=16 -> K={k+2,k+3}.
// D layout (ISA 32-bit C/D): VGPR i -> row M = i + (lane<16 ? 0 : 8), N = lane&15.
// ---------------------------------------------------------------------------
__global__ __launch_bounds__(256)
void seg_bev_wmma_kernel(const float* __restrict__ fg,
                         const int4* __restrict__ bounds,
                         float* __restrict__ bev) {
    __shared__ int4 sb[NA];
    const int b = blockIdx.y;
    stage_bounds_async(sb, bounds + b * NA);   // async DMA -> LDS, then barrier

    const int wave = threadIdx.x >> 5;
    const int lane = threadIdx.x & 31;
    const int tile = blockIdx.x * 8 + wave;   // 2500 blocks * 8 waves = 20000 tiles
    const int mtile = tile >> 3;              // 0..2499 (16-pixel tiles)
    const int cpair = tile & 7;               // 0..7    (32-channel tiles)
    const int pbase = mtile * 16;
    const int cbase = cpair * 32;

    // This lane's pixel (A operand) and two channels (B operands)
    const int p  = pbase + (lane & 15);
    const int py = p / W_;
    const int px = p % W_;
    const int khalf = (lane >> 4) << 1;       // lanes 16-31 handle K+2, K+3
    const int c0 = cbase + (lane & 15);
    const int c1 = c0 + 16;

    const float* fgb = fg + (size_t)b * NA * C_;

    v8f acc0 = {};
    v8f acc1 = {};
#pragma unroll 5
    for (int kk = 0; kk < NA; kk += 4) {
        const int k0 = kk + khalf;
        const int k1 = k0 + 1;
        const int4 r0 = sb[k0];               // ds_load_b128
        const int4 r1 = sb[k1];
        v2f a;
        a.x = (px >= r0.x && py >= r0.y && px < r0.z && py < r0.w) ? 1.0f : 0.0f;
        a.y = (px >= r1.x && py >= r1.y && px < r1.z && py < r1.w) ? 1.0f : 0.0f;
        const float* row0 = fgb + (size_t)k0 * C_;
        const float* row1 = fgb + (size_t)k1 * C_;
        v2f b0; b0.x = row0[c0]; b0.y = row1[c0];
        v2f b1; b1.x = row0[c1]; b1.y = row1[c1];
        // Two WMMAs share the same A operand (exact f32 accumulation)
        acc0 = __builtin_amdgcn_wmma_f32_16x16x4_f32(
            false, a, false, b0, (short)0, acc0, false, false);
        acc1 = __builtin_amdgcn_wmma_f32_16x16x4_f32(
            false, a, false, b1, (short)0, acc1, false, false);
    }

    // Store both 16x16 f32 D tiles (layout per ISA: VGPR i -> M = i or i+8)
    const int row0i = (lane >> 4) << 3;       // 0 or 8
    size_t base = (((size_t)b * HW_ + pbase + row0i) * C_) + (lane & 15);
#pragma unroll
    for (int i = 0; i < 8; ++i) {
        bev[base + (size_t)i * C_ + cbase]      = acc0[i];
        bev[base + (size_t)i * C_ + cbase + 16] = acc1[i];
    }
}

// ---------------------------------------------------------------------------
// Kernel 6: mask[b,h,w] = any box covers pixel (stored as 0.0/1.0 float)
// ---------------------------------------------------------------------------
__global__ void seg_mask_kernel(const int4* __restrict__ bounds,
                                float* __restrict__ mask) {
    __shared__ int4 sb[NA];
    const int b = blockIdx.y;
    stage_bounds_async(sb, bounds + b * NA);   // async DMA -> LDS, then barrier

    const int p = blockIdx.x * blockDim.x + threadIdx.x;
    if (p >= HW_) return;
    const int py = p / W_, px = p % W_;
    int any = 0;
#pragma unroll 4
    for (int k = 0; k < NA; ++k) {
        const int4 r = sb[k];
        any |= (px >= r.x && py >= r.y && px < r.z && py < r.w);
    }
    mask[(size_t)b * HW_ + p] = any ? 1.0f : 0.0f;
}

// ---------------------------------------------------------------------------
// Launch
// ---------------------------------------------------------------------------
extern "C" void kernel_launch(void* const* d_in, const int* in_sizes, int n_in,
                              void* d_out, int out_size, void* d_ws, size_t ws_size,
                              hipStream_t stream) {
    (void)in_sizes; (void)n_in; (void)out_size; (void)ws_size;

    const float* pred_boxes  = (const float*)d_in[0];  // [8,900,4]
    const float* pred_logits = (const float*)d_in[1];  // [8,900,11]
    const float* box_feats   = (const float*)d_in[2];  // [8,900,256]
    const float* view        = (const float*)d_in[3];  // [8,5,5]

    float* bev  = (float*)d_out;                       // [8,200,200,256] f32
    float* mask = bev + (size_t)B_ * HW_ * C_;         // [8,200,200] (0/1 as f32)

    // Workspace layout (~0.85 MB total; keep 16B alignment for int4 bounds)
    char*  ws      = (char*)d_ws;
    float* scores  = (float*)ws;                                   // 7200 f32
    int*   idx_sel = (int*)  (ws + (size_t)B_ * Q_ * 4);           // 800 i32
    int4*  bounds  = (int4*) (ws + (size_t)B_ * Q_ * 4
                                 + (size_t)B_ * NA * 4);           // 800 int4
    float* fg      = (float*)(ws + (size_t)B_ * Q_ * 4
                                 + (size_t)B_ * NA * 4
                                 + (size_t)B_ * NA * 16);          // 204800 f32

    seg_scores_kernel<<<(B_ * Q_ + 255) / 256, 256, 0, stream>>>(pred_logits, scores);
    seg_topk_kernel  <<<B_, 256, 0, stream>>>(scores, idx_sel);
    seg_decode_kernel<<<(B_ * NA + 255) / 256, 256, 0, stream>>>(pred_boxes, view,
                                                                 idx_sel, bounds);
    seg_gather_kernel<<<B_ * NA, C_, 0, stream>>>(box_feats, idx_sel, fg);

    // 2500 M-tiles * 8 (32-channel) C-tiles = 20000 tiles per batch; 8 waves/block
    seg_bev_wmma_kernel<<<dim3(2500, B_), 256, 0, stream>>>(fg, bounds, bev);
    seg_mask_kernel<<<dim3((HW_ + 255) / 256, B_), 256, 0, stream>>>(bounds, mask);
}